// DGCNN_TNet_75290776699035
// MI455X (gfx1250) — compile-verified
//
#include <hip/hip_runtime.h>
#include <hip/hip_bf16.h>
#include <stdint.h>

typedef __attribute__((ext_vector_type(2))) float v2f;
typedef __attribute__((ext_vector_type(8))) float v8f;

#define B_    8
#define N_    4096
#define KNN_  20
#define C1_   64
#define C2_   128
#define C3_   1024

// ---------------------------------------------------------------------------
// zero-init workspace (h2max + gmax are contiguous)
// ---------------------------------------------------------------------------
__global__ void k_init(unsigned* __restrict__ p, size_t n) {
  size_t i = (size_t)blockIdx.x * blockDim.x + threadIdx.x;
  size_t stride = (size_t)gridDim.x * blockDim.x;
  for (; i < n; i += stride) p[i] = 0u;
}

// ---------------------------------------------------------------------------
// KNN: one thread per (b, n). x[b] staged in LDS, top-20 kept in a guarded
// register insertion network (constant indices -> pure VGPRs, no scratch).
// ---------------------------------------------------------------------------
__global__ __launch_bounds__(256) void k_knn(const float* __restrict__ x,
                                             int* __restrict__ knn) {
  __shared__ float X[N_], Y[N_], Z[N_];
  const int b = blockIdx.x >> 4;
  const int n = ((blockIdx.x & 15) << 8) + threadIdx.x;
  const float* xb = x + (size_t)b * 3 * N_;
  for (int i = threadIdx.x; i < N_; i += 256) {
    X[i] = xb[i];
    Y[i] = xb[N_ + i];
    Z[i] = xb[2 * N_ + i];
  }
  __syncthreads();
  const float cx = X[n], cy = Y[n], cz = Z[n];
  float bd[KNN_];
  int   bix[KNN_];
#pragma unroll
  for (int j = 0; j < KNN_; ++j) { bd[j] = 3.0e38f; bix[j] = 0; }
  for (int m = 0; m < N_; ++m) {
    const float dx = X[m] - cx, dy = Y[m] - cy, dz = Z[m] - cz;
    const float d = dx * dx + dy * dy + dz * dz;
    if (d < bd[KNN_ - 1]) {            // amortized: ~k*ln(N/k) inserts per row
      float cd = d; int ci = m;
#pragma unroll
      for (int j = 0; j < KNN_; ++j) { // bubble-insert keeps ascending order
        if (cd < bd[j]) {
          float td = bd[j]; bd[j] = cd; cd = td;
          int   ti = bix[j]; bix[j] = ci; ci = ti;
        }
      }
    }
  }
  int* out = knn + ((size_t)b * N_ + n) * KNN_;
#pragma unroll
  for (int j = 0; j < KNN_; ++j) out[j] = bix[j];
}

// ---------------------------------------------------------------------------
// Edge conv: relu(W1@edge+b1) in VALU, relu(W2@h1+b2) on the WMMA pipe,
// max over k via float-bits atomicMax (values are >= 0 after relu).
// Block = 256 threads = 8 waves, 128 flat (n,k) columns per block.
// ---------------------------------------------------------------------------
__global__ __launch_bounds__(256) void k_edgeconv(
    const float* __restrict__ x, const int* __restrict__ knn,
    const float* __restrict__ W1, const float* __restrict__ b1,
    const float* __restrict__ W2, const float* __restrict__ b2,
    unsigned* __restrict__ h2max) {
  __shared__ float eLDS[128][6];          // edge features for 128 columns
  __shared__ float h1LDS[8][C1_][16];     // per-wave h1 tile, [K][N] layout

  const int t   = threadIdx.x;
  const int b   = blockIdx.x / 640;
  const int blk = blockIdx.x % 640;
  const int f0  = blk * 128;              // flat (n,k) base for this block
  const float* xb = x + (size_t)b * 3 * N_;

  // ---- phase 0: build edge features (threads 0..127, one column each) ----
  if (t < 128) {
    const int f = f0 + t;
    const int n = f / KNN_;
    const int k = f - n * KNN_;
    const int j = knn[((size_t)b * N_ + n) * KNN_ + k];
    const float cx = xb[n], cy = xb[N_ + n], cz = xb[2 * N_ + n];
    eLDS[t][0] = cx;
    eLDS[t][1] = cy;
    eLDS[t][2] = cz;
    eLDS[t][3] = xb[j] - cx;
    eLDS[t][4] = xb[N_ + j] - cy;
    eLDS[t][5] = xb[2 * N_ + j] - cz;
  }
  __syncthreads();

  // ---- phase 1: h1 = relu(W1@edge + b1); 32 values per thread ----
  {
    const int col  = t & 127;
    const int rb   = (t >> 7) * 32;       // rows 0..31 or 32..63
    const int tile = col >> 4, c = col & 15;
    const float e0 = eLDS[col][0], e1 = eLDS[col][1], e2 = eLDS[col][2];
    const float e3 = eLDS[col][3], e4 = eLDS[col][4], e5 = eLDS[col][5];
    for (int r = rb; r < rb + 32; ++r) {
      const float* w = W1 + r * 6;
      float a = b1[r] + w[0] * e0 + w[1] * e1 + w[2] * e2 +
                        w[3] * e3 + w[4] * e4 + w[5] * e5;
      h1LDS[tile][r][c] = fmaxf(a, 0.f);
    }
  }
  __syncthreads();

  // ---- phase 2: f32 WMMA, one 16-column tile per wave ----
  const int wave = t >> 5, lane = t & 31;
  const int half = lane >> 4, lc = lane & 15;
  const float(*h1)[16] = h1LDS[wave];
  const int fbase = f0 + wave * 16;

  for (int tm = 0; tm < 8; ++tm) {        // 8 M-tiles cover 128 output rows
    v8f acc = {};
    const int arow = tm * 16 + lc;        // A fragment: M = lane[3:0]
#pragma unroll
    for (int s = 0; s < 16; ++s) {        // K = 64 in steps of 4
      v2f a, bb;
      a.x  = W2[arow * 64 + s * 4 + 2 * half];       // K = 0 / 2
      a.y  = W2[arow * 64 + s * 4 + 2 * half + 1];   // K = 1 / 3
      bb.x = h1[s * 4 + 2 * half][lc];               // B: N = lane[3:0]
      bb.y = h1[s * 4 + 2 * half + 1][lc];
      acc = __builtin_amdgcn_wmma_f32_16x16x4_f32(
          false, a, false, bb, (short)0, acc, false, false);
    }
    const int f = fbase + lc;
    const int n = f / KNN_;
#pragma unroll
    for (int r = 0; r < 8; ++r) {         // D layout: M = r + 8*half, N = lc
      const int m = tm * 16 + r + half * 8;
      const float v = fmaxf(acc[r] + b2[m], 0.f);
      atomicMax(&h2max[((size_t)b * C2_ + m) * N_ + n], __float_as_uint(v));
    }
  }
}

// ---------------------------------------------------------------------------
// Point MLP: relu(Wi@h2max + bi) on WMMA, fused max over N.
// Grid = B * 8 row-slices (128 Wi rows each); block = 8 waves, each wave owns
// one 16-row M-tile; column tiles of 16 points staged through LDS.
// ---------------------------------------------------------------------------
__global__ __launch_bounds__(256) void k_point_mlp(
    const float* __restrict__ h2maxF, const float* __restrict__ Wi,
    const float* __restrict__ bi, unsigned* __restrict__ gmax) {
  __shared__ float    h2t[C2_][16];
  __shared__ unsigned gpart[C2_];
  const int t = threadIdx.x;
  const int b = blockIdx.x >> 3;
  const int slice = blockIdx.x & 7;
  if (t < C2_) gpart[t] = 0u;
  const int wave = t >> 5, lane = t & 31, half = lane >> 4, lc = lane & 15;
  const int mbase = slice * 128 + wave * 16;
  const float* h2b = h2maxF + (size_t)b * C2_ * N_;
  __syncthreads();

  for (int ct = 0; ct < N_ / 16; ++ct) {
    // stage one 128x16 input tile
#pragma unroll
    for (int i = 0; i < 8; ++i) {
      const int e = t + 256 * i;
      const int row = e >> 4, c = e & 15;
      h2t[row][c] = h2b[(size_t)row * N_ + ct * 16 + c];
    }
    __syncthreads();

    v8f acc = {};
    const int arow = mbase + lc;
#pragma unroll
    for (int s = 0; s < 32; ++s) {        // K = 128 in steps of 4
      v2f a, bb;
      a.x  = Wi[(size_t)arow * 128 + s * 4 + 2 * half];
      a.y  = Wi[(size_t)arow * 128 + s * 4 + 2 * half + 1];
      bb.x = h2t[s * 4 + 2 * half][lc];
      bb.y = h2t[s * 4 + 2 * half + 1][lc];
      acc = __builtin_amdgcn_wmma_f32_16x16x4_f32(
          false, a, false, bb, (short)0, acc, false, false);
    }
#pragma unroll
    for (int r = 0; r < 8; ++r) {
      const int mloc = wave * 16 + r + half * 8;
      const float v = fmaxf(acc[r] + bi[slice * 128 + mloc], 0.f);
      atomicMax(&gpart[mloc], __float_as_uint(v));   // ds_max_u32
    }
    __syncthreads();
  }
  if (t < C2_) atomicMax(&gmax[b * 1024 + slice * 128 + t], gpart[t]);
}

// ---------------------------------------------------------------------------
// FC head: 1024 -> 512 -> 256 -> 9 (+ identity). Tiny; VALU + LDS.
// ---------------------------------------------------------------------------
__global__ __launch_bounds__(256) void k_head(
    const float* __restrict__ gmaxF,
    const float* __restrict__ Wg1, const float* __restrict__ bg1,
    const float* __restrict__ Wg2, const float* __restrict__ bg2,
    const float* __restrict__ Wl,  const float* __restrict__ bl,
    float* __restrict__ out) {
  __shared__ float g[1024], g1[512], g2[256];
  const int t = threadIdx.x, b = blockIdx.x;
  for (int i = t; i < 1024; i += 256) g[i] = gmaxF[b * 1024 + i];
  __syncthreads();
  for (int r = t; r < 512; r += 256) {
    float acc = bg1[r];
    const float* w = Wg1 + (size_t)r * 1024;
    for (int q = 0; q < 1024; ++q) acc += w[q] * g[q];
    g1[r] = fmaxf(acc, 0.f);
  }
  __syncthreads();
  {
    float acc = bg2[t];
    const float* w = Wg2 + (size_t)t * 512;
    for (int q = 0; q < 512; ++q) acc += w[q] * g1[q];
    g2[t] = fmaxf(acc, 0.f);
  }
  __syncthreads();
  if (t < 9) {
    float acc = bl[t];
    const float* w = Wl + t * 256;
    for (int q = 0; q < 256; ++q) acc += w[q] * g2[q];
    if (t == 0 || t == 4 || t == 8) acc += 1.0f;   // + eye(3)
    out[b * 9 + t] = acc;
  }
}

// ---------------------------------------------------------------------------
extern "C" void kernel_launch(void* const* d_in, const int* in_sizes, int n_in,
                              void* d_out, int out_size, void* d_ws,
                              size_t ws_size, hipStream_t stream) {
  const float* x   = (const float*)d_in[0];
  const float* W1  = (const float*)d_in[1];
  const float* b1  = (const float*)d_in[2];
  const float* W2  = (const float*)d_in[3];
  const float* b2  = (const float*)d_in[4];
  const float* Wi  = (const float*)d_in[5];
  const float* bi  = (const float*)d_in[6];
  const float* Wg1 = (const float*)d_in[7];
  const float* bg1 = (const float*)d_in[8];
  const float* Wg2 = (const float*)d_in[9];
  const float* bg2 = (const float*)d_in[10];
  const float* Wl  = (const float*)d_in[11];
  const float* bl  = (const float*)d_in[12];

  char* ws = (char*)d_ws;
  int*      knn   = (int*)ws;                              // B*N*20 ints
  const size_t knnBytes = (size_t)B_ * N_ * KNN_ * sizeof(int);
  unsigned* h2max = (unsigned*)(ws + knnBytes);            // B*128*N (f32 bits)
  const size_t h2Bytes = (size_t)B_ * C2_ * N_ * sizeof(unsigned);
  unsigned* gmax  = (unsigned*)(ws + knnBytes + h2Bytes);  // B*1024 (f32 bits)

  const size_t zeroN = (size_t)B_ * C2_ * N_ + (size_t)B_ * 1024;
  k_init<<<2048, 256, 0, stream>>>(h2max, zeroN);          // h2max + gmax
  k_knn<<<B_ * (N_ / 256), 256, 0, stream>>>(x, knn);
  k_edgeconv<<<B_ * 640, 256, 0, stream>>>(x, knn, W1, b1, W2, b2, h2max);
  k_point_mlp<<<B_ * 8, 256, 0, stream>>>((const float*)h2max, Wi, bi, gmax);
  k_head<<<B_, 256, 0, stream>>>((const float*)gmax, Wg1, bg1, Wg2, bg2, Wl, bl,
                                 (float*)d_out);
}